// ModelSGLang_68186900792187
// MI455X (gfx1250) — compile-verified
//
#include <hip/hip_runtime.h>

// 128-bit vector of int32 for b128 global memory ops.
typedef int v4i __attribute__((ext_vector_type(4)));

// Grid: (k, bs). Row = blockIdx.y*k + blockIdx.x; its A-source row is blockIdx.y.
// Row layout (int32):
//   pos <  sa           : page_table_a[blockIdx.y][pos]
//   sa <= pos < sa+sb   : page_table_b[row][pos-sa]
//   pos >= sa+sb        : page_table_dst[row][pos]  (passthrough)
__global__ __launch_bounds__(256) void concat_page_table_kernel(
    const int* __restrict__ dst_in,   // [bs_expand, dst_len]
    const int* __restrict__ pa,       // [bs, len_a]
    const int* __restrict__ pb,       // [bs_expand, len_b]
    const int* __restrict__ sla,      // [bs]
    const int* __restrict__ slb,      // [bs_expand]
    int* __restrict__ out,            // [bs_expand, dst_len]
    int len_a, int len_b, int dst_len)
{
    const int arow_idx = (int)blockIdx.y;
    const int row      = arow_idx * (int)gridDim.x + (int)blockIdx.x;

    const int sa = sla[arow_idx];          // block-uniform -> SGPR
    const int sb = slb[row];               // block-uniform -> SGPR
    const int se = sa + sb;

    const int* __restrict__ arow = pa     + (size_t)arow_idx * (size_t)len_a;
    const int* __restrict__ brow = pb     + (size_t)row      * (size_t)len_b;
    const int* __restrict__ drow = dst_in + (size_t)row      * (size_t)dst_len;
    int*       __restrict__ orow = out    + (size_t)row      * (size_t)dst_len;

    const int nvec   = dst_len >> 2;       // 4112 chunks for dst_len=16448
    const int stride = (int)blockDim.x;    // 256 threads = 8 wave32 waves
    const int tid    = (int)threadIdx.x;

    // Scalar (block-uniform) range split: [0,aend) all-A, [aend,dstart) mixed,
    // [dstart,nvec) all-dst. dst_len % 4 == 0 here, but a scalar tail below
    // keeps the kernel generic.
    const int aend_raw   = sa >> 2;
    const int aend       = aend_raw < nvec ? aend_raw : nvec;
    const int dstart_raw = (se + 3) >> 2;
    const int dstart     = dstart_raw < nvec ? dstart_raw : nvec;

    // ---- Bulk stream 1: A -> out (branch-free b128 copy) ----
    for (int c = tid; c < aend; c += stride) {
        const int p0 = c << 2;
        __builtin_prefetch(arow + p0 + 4 * stride, 0, 0);   // global_prefetch_b8
        v4i v = *(const v4i*)(arow + p0);
        __builtin_nontemporal_store(v, (v4i*)(orow + p0));  // NT: touch-once
    }

    // ---- Mixed / B window: <= ~18 chunks per row, scalar gather ----
    for (int c = aend + tid; c < dstart; c += stride) {
        const int p0 = c << 2;
        int r[4];
#pragma unroll
        for (int j = 0; j < 4; ++j) {
            const int pos = p0 + j;
            int val;
            if (pos < sa)      val = arow[pos];
            else if (pos < se) val = brow[pos - sa];
            else               val = drow[pos];
            r[j] = val;
        }
        v4i v = { r[0], r[1], r[2], r[3] };
        __builtin_nontemporal_store(v, (v4i*)(orow + p0));
    }

    // ---- Bulk stream 2: dst passthrough (branch-free NT b128 copy) ----
    for (int c = dstart + tid; c < nvec; c += stride) {
        const int p0 = c << 2;
        __builtin_prefetch(drow + p0 + 4 * stride, 0, 0);   // global_prefetch_b8
        v4i v = __builtin_nontemporal_load((const v4i*)(drow + p0));
        __builtin_nontemporal_store(v, (v4i*)(orow + p0));
    }

    // Generic scalar tail if dst_len % 4 != 0 (not hit for 16448).
    for (int pos = (nvec << 2) + tid; pos < dst_len; pos += stride) {
        int val;
        if (pos < sa)      val = arow[pos];
        else if (pos < se) val = brow[pos - sa];
        else               val = drow[pos];
        orow[pos] = val;
    }
}

extern "C" void kernel_launch(void* const* d_in, const int* in_sizes, int n_in,
                              void* d_out, int out_size, void* d_ws, size_t ws_size,
                              hipStream_t stream) {
    // setup_inputs() order:
    //  0: page_table_dst [bs_expand, dst_len]  int32
    //  1: page_table_a   [bs, len_a]           int32
    //  2: page_table_b   [bs_expand, len_b]    int32
    //  3: seq_len_a      [bs]                  int32
    //  4: seq_len_b      [bs_expand]           int32
    const int* dst_in = (const int*)d_in[0];
    const int* pa     = (const int*)d_in[1];
    const int* pb     = (const int*)d_in[2];
    const int* sla    = (const int*)d_in[3];
    const int* slb    = (const int*)d_in[4];

    const int bs        = in_sizes[3];
    const int bs_expand = in_sizes[4];
    const int k         = bs_expand / bs;
    const int len_a     = in_sizes[1] / bs;
    const int len_b     = in_sizes[2] / bs_expand;
    const int dst_len   = in_sizes[0] / bs_expand;

    int* out = (int*)d_out;  // int32 bit patterns through the output buffer

    dim3 grid((unsigned)k, (unsigned)bs);   // blockIdx.y = A row, blockIdx.x = draft slot
    dim3 block(256);
    hipLaunchKernelGGL(concat_page_table_kernel, grid, block, 0, stream,
                       dst_in, pa, pb, sla, slb, out,
                       len_a, len_b, dst_len);
}